// Attention_51032801411515
// MI455X (gfx1250) — compile-verified
//
#include <hip/hip_runtime.h>
#include <math.h>

#define B_    512
#define N_    600
#define H_    256
#define K_    9
#define WIN_  129
#define PAD_  4
#define PWIN_ 137   // WIN + 2*PAD
#define NP_   608   // N + 2*PAD
#define AVG_  1.5f

typedef __attribute__((ext_vector_type(2))) float v2f;
typedef __attribute__((ext_vector_type(8))) float v8f;

// ---------------------------------------------------------------------------
// Kernel A: qw[b, j] = sum_h query[b,h] * conv_w[h, j],  j in [0,18)
// GEMM (512x256) x (256x18) using V_WMMA_F32_16X16X4_F32.
// conv_w is staged once into LDS zero-padded to 32 columns so the K-loop is a
// branch-free ds_load + wmma stream (no per-iteration exec-mask guards).
// ---------------------------------------------------------------------------
__global__ __launch_bounds__(32) void qw_wmma_kernel(const float* __restrict__ q,
                                                     const float* __restrict__ w,
                                                     float* __restrict__ qw) {
  __shared__ float Wp[H_ * 32];       // 256 x 32, cols 18..31 = 0

  const int lane = threadIdx.x;       // 0..31
  const int m    = lane & 15;         // A row / D column
  const int hi   = lane >> 4;         // 0 -> K pair {0,1}, 1 -> K pair {2,3}
  const int row0 = blockIdx.x * 16;   // batch tile base

  // Cooperative zero-padded stage of conv_w (256 x 18) -> LDS (256 x 32).
  for (int idx = lane; idx < H_ * 32; idx += 32) {
    const int k = idx >> 5;
    const int j = idx & 31;
    Wp[idx] = (j < 18) ? w[k * 18 + j] : 0.0f;
  }
  __syncthreads();

  const float* qrow = q + (row0 + m) * H_;

  v8f acc0 = {};
  v8f acc1 = {};
  for (int k0 = 0; k0 < H_; k0 += 4) {
    const int ka = k0 + 2 * hi;
    v2f a;
    a.x = qrow[ka];
    a.y = qrow[ka + 1];
    v2f b0, b1;
    b0.x = Wp[ka * 32 + m];
    b0.y = Wp[(ka + 1) * 32 + m];
    b1.x = Wp[ka * 32 + 16 + m];
    b1.y = Wp[(ka + 1) * 32 + 16 + m];
    acc0 = __builtin_amdgcn_wmma_f32_16x16x4_f32(false, a, false, b0, (short)0, acc0, false, false);
    acc1 = __builtin_amdgcn_wmma_f32_16x16x4_f32(false, a, false, b1, (short)0, acc1, false, false);
  }
  // D layout: VGPR v holds row (v + 8*hi), column = lane & 15
  for (int v = 0; v < 8; ++v) {
    const int r = row0 + v + 8 * hi;
    qw[r * 18 + m] = acc0[v];
    const int j1 = 16 + m;
    if (j1 < 18) qw[r * 18 + j1] = acc1[v];
  }
}

// ---------------------------------------------------------------------------
// Kernel Z: zero unpadded & full_align regions, copy cum_alignment -> cum_out.
// ---------------------------------------------------------------------------
__global__ void init_kernel(const float* __restrict__ cum_in,
                            float* __restrict__ unpadded,
                            float* __restrict__ full_align,
                            float* __restrict__ cum_out) {
  const int i = blockIdx.x * blockDim.x + threadIdx.x;
  if (i < B_ * NP_) {
    full_align[i] = 0.0f;
    cum_out[i]    = cum_in[i];
  }
  if (i < B_ * N_) unpadded[i] = 0.0f;
}

// ---------------------------------------------------------------------------
// Kernel B: one block (256 threads = 8 wave32) per batch element.
// Fused: position features (from LDS) folded through qw, score GEMV over H,
// masked softmax, context GEMV over WIN, scatter, tie-breaking argmax.
// Memory bound: ~140 MB total traffic -> ~6 us floor at 23.3 TB/s.
// ---------------------------------------------------------------------------
__global__ __launch_bounds__(256) void attn_kernel(
    const float* __restrict__ tokens,      // (B,N,H)
    const float* __restrict__ token_keys,  // (B,H,N)
    const int*   __restrict__ num_tokens,  // (B)
    const float* __restrict__ query,       // (B,H)
    const float* __restrict__ algn,        // (B,NP)
    const float* __restrict__ cum_in,      // (B,NP)
    const int*   __restrict__ wstart,      // (B)
    const float* __restrict__ conv_b,      // (H)
    const float* __restrict__ qw,          // (B,18) from kernel A
    float* __restrict__ ctx,               // (B,H)
    float* __restrict__ unpadded,          // (B,N)
    float* __restrict__ full_align,        // (B,NP)
    float* __restrict__ cum_out,           // (B,NP)
    int*   __restrict__ new_start)         // (B)
{
  __shared__ float pos0[PWIN_];
  __shared__ float pos1[PWIN_];
  __shared__ float qs[H_];
  __shared__ float qws[18];
  __shared__ float als[256];
  __shared__ float red[256];
  __shared__ int   ridx[256];

  const int b  = blockIdx.x;
  const int t  = threadIdx.x;
  const int ws = wstart[b];
  const int nt = num_tokens[b];

  qs[t] = query[b * H_ + t];
  if (t < PWIN_) {
    pos0[t] = cum_in[b * NP_ + ws + t] * (1.0f / AVG_) - 1.0f;  // cum/AVG - 1
    pos1[t] = algn[b * NP_ + ws + t] * 2.0f - 1.0f;             // last*2 - 1
  }
  if (t < 18) qws[t] = qw[b * 18 + t];
  __syncthreads();

  // qb = sum_h q[h] * conv_b[h]  (conv bias folded through query)
  red[t] = qs[t] * conv_b[t];
  __syncthreads();
  for (int s = 128; s > 0; s >>= 1) {
    if (t < s) red[t] += red[t + s];
    __syncthreads();
  }
  const float qb = red[0];
  __syncthreads();

  // score[w]: thread t = w position (coalesced token_keys reads along w)
  float sc = -INFINITY;
  if (t < WIN_) {
    float qk = 0.0f;
    const float* kp = token_keys + (size_t)b * H_ * N_ + ws + t;
#pragma unroll 8
    for (int h = 0; h < H_; ++h) qk += qs[h] * kp[h * N_];
    float cv = qb;
#pragma unroll
    for (int k = 0; k < K_; ++k)
      cv += pos0[t + k] * qws[k] + pos1[t + k] * qws[K_ + k];
    sc = (qk + cv) * (1.0f / 16.0f);   // / sqrt(H), H = 256
    if (ws + t >= nt) sc = -INFINITY;  // tokens_mask recomputed from num_tokens
  }

  // softmax: block max
  red[t] = sc;
  __syncthreads();
  for (int s = 128; s > 0; s >>= 1) {
    if (t < s) red[t] = fmaxf(red[t], red[t + s]);
    __syncthreads();
  }
  const float mx = red[0];
  __syncthreads();
  const float e = (t < WIN_) ? __expf(sc - mx) : 0.0f;
  red[t] = e;
  __syncthreads();
  for (int s = 128; s > 0; s >>= 1) {
    if (t < s) red[t] += red[t + s];
    __syncthreads();
  }
  const float inv = 1.0f / red[0];
  __syncthreads();
  const float a = e * inv;
  als[t] = (t < WIN_) ? a : 0.0f;
  __syncthreads();

  // context[b,h]: thread t = h (coalesced tokens reads along h)
  {
    float acc = 0.0f;
    const float* tp = tokens + (size_t)b * N_ * H_ + (size_t)ws * H_ + t;
#pragma unroll 4
    for (int w = 0; w < WIN_; ++w) acc += als[w] * tp[w * H_];
    ctx[b * H_ + t] = acc;
  }

  // scatter alignment outputs
  if (t < WIN_) {
    const int p = ws + t;
    const float av = als[t];
    unpadded[b * N_ + p]            = av;                              // full_align[:, PAD:-PAD]
    full_align[b * NP_ + p + PAD_]  = av;
    cum_out[b * NP_ + p + PAD_]     = cum_in[b * NP_ + p + PAD_] + av; // cum + full_align
  }

  // argmax over full_align: softmax values > 0 so max lies in window;
  // tie-break = smallest index (jnp.argmax semantics).
  red[t]  = (t < WIN_) ? als[t] : -1.0f;
  ridx[t] = PAD_ + ws + t;
  __syncthreads();
  for (int s = 128; s > 0; s >>= 1) {
    if (t < s) {
      const float v2 = red[t + s];
      const int   i2 = ridx[t + s];
      if (v2 > red[t] || (v2 == red[t] && i2 < ridx[t])) {
        red[t] = v2;
        ridx[t] = i2;
      }
    }
    __syncthreads();
  }
  if (t == 0) {
    int nsv = ridx[0] - (PWIN_ / 2);   // argmax - 68
    nsv = max(ws, nsv);
    nsv = min(nsv, nt - WIN_);
    nsv = max(nsv, 0);
    new_start[b] = nsv;
  }
}

// ---------------------------------------------------------------------------
extern "C" void kernel_launch(void* const* d_in, const int* in_sizes, int n_in,
                              void* d_out, int out_size, void* d_ws, size_t ws_size,
                              hipStream_t stream) {
  (void)in_sizes; (void)n_in; (void)out_size; (void)ws_size;
  const float* tokens = (const float*)d_in[0];
  // d_in[1] = tokens_mask: unused (recomputed from num_tokens)
  const float* tkeys  = (const float*)d_in[2];
  const int*   ntok   = (const int*)d_in[3];
  const float* query  = (const float*)d_in[4];
  const float* algn   = (const float*)d_in[5];
  const float* cum    = (const float*)d_in[6];
  const int*   wstart = (const int*)d_in[7];
  const float* conv_w = (const float*)d_in[8];
  const float* conv_b = (const float*)d_in[9];

  // Output layout (flat, return order): context | unpadded | full_align | cum_out | new_start
  float* out       = (float*)d_out;
  float* ctx       = out;
  float* unpadded  = ctx + B_ * H_;
  float* fal       = unpadded + B_ * N_;
  float* cum_out   = fal + B_ * NP_;
  int*   new_start = (int*)(cum_out + B_ * NP_);

  float* qw = (float*)d_ws;  // B*18 floats of scratch

  qw_wmma_kernel<<<B_ / 16, 32, 0, stream>>>(query, conv_w, qw);
  init_kernel<<<(B_ * NP_ + 255) / 256, 256, 0, stream>>>(cum, unpadded, fal, cum_out);
  attn_kernel<<<B_, 256, 0, stream>>>(tokens, tkeys, ntok, query, algn, cum, wstart,
                                      conv_b, qw, ctx, unpadded, fal, cum_out, new_start);
}